// TwoLayerBidirectionalLstmModel_61452392071467
// MI455X (gfx1250) — compile-verified
//
#include <hip/hip_runtime.h>

// Problem constants (from reference): S=512, B=8192, I=3, H=5
#define SS 512
#define BB 8192
#define HH 5
#define GATES 20      // 4*H
#define KPAD 16       // A operand K: [h(5) | x(Din<=10) | pad | 1.0 at k=15]
#define ASTRIDE 18    // ldsA row stride (even -> b64-aligned, reduces bank conflicts)
#define GSTRIDE 33    // ldsG row stride (odd -> no 2-way conflict on scatter)

typedef float v2f __attribute__((ext_vector_type(2)));
typedef float v8f __attribute__((ext_vector_type(8)));

// fast activations: single v_exp_f32 + v_rcp_f32, no IEEE-div fixup chain
__device__ __forceinline__ float fast_sigmoid(float z) {
    return __builtin_amdgcn_rcpf(1.0f + __expf(-z));
}
__device__ __forceinline__ float fast_tanh(float z) {
    // tanh(z) = 1 - 2/(exp(2z)+1)
    return 1.0f - 2.0f * __builtin_amdgcn_rcpf(1.0f + __expf(2.0f * z));
}

// One wave (32 threads) scans a 16-row batch tile over all 512 steps for one
// direction.  Per step:  gates[16x20] = [h | x | 1] (16x16) * Wc (16x32)
// via 8 chained v_wmma_f32_16x16x4_f32 (full f32, matches reference precision).
// x(s+1) is software-pipelined: global loads issued under the WMMA chain,
// committed to the LDS A-staging tile during the activation phase.
template <int DIN>
__global__ __launch_bounds__(32) void lstm_dir_scan(
    const float* __restrict__ in,                                   // [S,B,DIN]
    const float* __restrict__ wih0, const float* __restrict__ whh0, // fwd weights
    const float* __restrict__ bih0, const float* __restrict__ bhh0,
    const float* __restrict__ wih1, const float* __restrict__ whh1, // bwd weights
    const float* __restrict__ bih1, const float* __restrict__ bhh1,
    float* __restrict__ hs_out,                                     // [S,B,2H]
    float* __restrict__ hn, float* __restrict__ cn,                 // [4,B,H]
    int dirBase)
{
    constexpr int NX = (16 * DIN + 31) / 32;  // x-slab elements per lane

    const int lane  = threadIdx.x;      // 0..31
    const int bbase = blockIdx.x * 16;  // batch tile base
    const int dir   = blockIdx.y;       // 0 = fwd, 1 = bwd
    const int dirIdx = dirBase + dir;

    const float* wih = dir ? wih1 : wih0;
    const float* whh = dir ? whh1 : whh0;
    const float* bih = dir ? bih1 : bih0;
    const float* bhh = dir ? bhh1 : bhh0;

    __shared__ float ldsA[16][ASTRIDE];  // A staging, k-major per row: [row][k]
    __shared__ float ldsG[16][GSTRIDE];  // gate tile after WMMA: [row][gate]

    // init A staging: zeros, k=15 slot carries the bias multiplier 1.0
    for (int e = lane; e < 16 * KPAD; e += 32) {
        int r = e >> 4, k = e & 15;
        ldsA[r][k] = (k == 15) ? 1.0f : 0.0f;
    }

    const int halt = lane >> 4;   // which 16-lane half
    const int nlo  = lane & 15;
    const int arow = nlo;         // A-matrix row owned by this lane

    // Loop-invariant B operand: Wc[k][n]  (k = [w_hh | w_ih | pad | bias-row])
    // Layout for 16x16x4 f32 operands: VGPR v, lane half h -> K = 2h+v, N = lane&15.
    v2f Bt[2][4];
    for (int t = 0; t < 2; ++t) {
        int n = nlo + 16 * t;
        for (int kc = 0; kc < 4; ++kc) {
            for (int v = 0; v < 2; ++v) {
                int k = 4 * kc + 2 * halt + v;
                float w = 0.0f;
                if (n < GATES) {
                    if (k < HH)            w = whh[n * HH + k];
                    else if (k < HH + DIN) w = wih[n * DIN + (k - HH)];
                    else if (k == 15)      w = bih[n] + bhh[n];
                }
                Bt[t][kc][v] = w;
            }
        }
    }

    // activation lanes: pair p = lane + 32*q -> (row = p&15, hidden j = p>>4)
    float c_st[3] = {0.f, 0.f, 0.f};
    float h_st[3] = {0.f, 0.f, 0.f};

    // ---- prologue: stage x(s0) ----
    {
        const int s0 = dir ? (SS - 1) : 0;
        const float* src = in + ((size_t)s0 * BB + bbase) * DIN;
#pragma unroll
        for (int i = 0; i < NX; ++i) {
            int e = lane + 32 * i;
            if (e < 16 * DIN) ldsA[e / DIN][HH + e % DIN] = src[e];
        }
    }
    __syncthreads();

    for (int it = 0; it < SS; ++it) {
        const int s = dir ? (SS - 1 - it) : it;

        // ---- gather A chunks: two consecutive k per lane -> b64 LDS loads ----
        v2f a[4];
#pragma unroll
        for (int kc = 0; kc < 4; ++kc) {
            a[kc] = *reinterpret_cast<const v2f*>(&ldsA[arow][4 * kc + 2 * halt]);
        }

        // ---- pipeline: issue x(s+1) loads now, latency hidden by WMMA+act ----
        float xr[NX];
        const bool have_next = (it + 1 < SS);
        if (have_next) {
            const int s2 = dir ? (s - 1) : (s + 1);
            const float* src = in + ((size_t)s2 * BB + bbase) * DIN;
#pragma unroll
            for (int i = 0; i < NX; ++i) {
                int e = lane + 32 * i;
                if (e < 16 * DIN) xr[i] = src[e];
            }
        }
        if (it + 2 < SS) {  // warm L0/L2 for step s+2 (gfx1250 global_prefetch_b8)
            const int s3 = dir ? (s - 2) : (s + 2);
            const char* pf = (const char*)(in + ((size_t)s3 * BB + bbase) * DIN);
            if (lane * 64 < 16 * DIN * 4) __builtin_prefetch(pf + lane * 64, 0, 0);
        }

        // ---- fused recurrent+input matmul+bias: 8 chained WMMAs ----
        v8f c0 = {};
        v8f c1 = {};
#pragma unroll
        for (int kc = 0; kc < 4; ++kc) {
            c0 = __builtin_amdgcn_wmma_f32_16x16x4_f32(false, a[kc], false, Bt[0][kc],
                                                       (short)0, c0, false, false);
            c1 = __builtin_amdgcn_wmma_f32_16x16x4_f32(false, a[kc], false, Bt[1][kc],
                                                       (short)0, c1, false, false);
        }

        // ---- scatter gate tile to LDS (C layout: VGPR j -> row j+8*half, col lane&15) ----
#pragma unroll
        for (int j = 0; j < 8; ++j) {
            int r = j + 8 * halt;
            ldsG[r][nlo]      = c0[j];
            ldsG[r][nlo + 16] = c1[j];
        }
        __syncthreads();

        // ---- activations + state update: 80 (row,hidden) pairs over 32 lanes ----
        auto act = [&](int q, int p) {
            int r = p & 15, j = p >> 4;
            float gi = ldsG[r][j];
            float gf = ldsG[r][HH + j];
            float gg = ldsG[r][2 * HH + j];
            float go = ldsG[r][3 * HH + j];
            float cc = fast_sigmoid(gf) * c_st[q] + fast_sigmoid(gi) * fast_tanh(gg);
            float hh = fast_sigmoid(go) * fast_tanh(cc);
            c_st[q] = cc;
            h_st[q] = hh;
            ldsA[r][j] = hh;  // feed h into next step's A operand
            hs_out[((size_t)s * BB + (bbase + r)) * (2 * HH) + dir * HH + j] = hh;
        };
        act(0, lane);                       // p = 0..31   always valid
        act(1, lane + 32);                  // p = 32..63  always valid
        if (lane < 16) act(2, lane + 64);   // p = 64..79

        // ---- commit pipelined x(s+1) into the A-staging tile ----
        if (have_next) {
#pragma unroll
            for (int i = 0; i < NX; ++i) {
                int e = lane + 32 * i;
                if (e < 16 * DIN) ldsA[e / DIN][HH + e % DIN] = xr[i];
            }
        }
        __syncthreads();
    }

    // ---- final h_T / c_T ----
    auto fin = [&](int q, int p) {
        int r = p & 15, j = p >> 4;
        size_t idx = ((size_t)dirIdx * BB + (bbase + r)) * HH + j;
        hn[idx] = h_st[q];
        cn[idx] = c_st[q];
    };
    fin(0, lane);
    fin(1, lane + 32);
    if (lane < 16) fin(2, lane + 64);
}

extern "C" void kernel_launch(void* const* d_in, const int* in_sizes, int n_in,
                              void* d_out, int out_size, void* d_ws, size_t ws_size,
                              hipStream_t stream) {
    (void)in_sizes; (void)n_in; (void)out_size; (void)ws_size;

    const float* x        = (const float*)d_in[0];
    const float* wih_l0   = (const float*)d_in[1];
    const float* whh_l0   = (const float*)d_in[2];
    const float* bih_l0   = (const float*)d_in[3];
    const float* bhh_l0   = (const float*)d_in[4];
    const float* wih_l0r  = (const float*)d_in[5];
    const float* whh_l0r  = (const float*)d_in[6];
    const float* bih_l0r  = (const float*)d_in[7];
    const float* bhh_l0r  = (const float*)d_in[8];
    const float* wih_l1   = (const float*)d_in[9];
    const float* whh_l1   = (const float*)d_in[10];
    const float* bih_l1   = (const float*)d_in[11];
    const float* bhh_l1   = (const float*)d_in[12];
    const float* wih_l1r  = (const float*)d_in[13];
    const float* whh_l1r  = (const float*)d_in[14];
    const float* bih_l1r  = (const float*)d_in[15];
    const float* bhh_l1r  = (const float*)d_in[16];

    float* out  = (float*)d_out;                    // [S,B,10]
    float* hn   = out + (size_t)SS * BB * 2 * HH;   // [4,B,5]
    float* cn   = hn + (size_t)4 * BB * HH;         // [4,B,5]
    float* out0 = (float*)d_ws;                     // layer-0 output [S,B,10] = 168 MB

    dim3 grid(BB / 16, 2);
    dim3 block(32);

    // layer 0 (Din = 3), both directions in parallel across grid.y
    lstm_dir_scan<3><<<grid, block, 0, stream>>>(
        x, wih_l0, whh_l0, bih_l0, bhh_l0, wih_l0r, whh_l0r, bih_l0r, bhh_l0r,
        out0, hn, cn, 0);

    // layer 1 (Din = 10) reads layer-0 output
    lstm_dir_scan<10><<<grid, block, 0, stream>>>(
        out0, wih_l1, whh_l1, bih_l1, bhh_l1, wih_l1r, whh_l1r, bih_l1r, bhh_l1r,
        out, hn, cn, 2);
}